// GNN_68453188764137
// MI455X (gfx1250) — compile-verified
//
#include <hip/hip_runtime.h>

typedef float v2f __attribute__((ext_vector_type(2)));
typedef float v8f __attribute__((ext_vector_type(8)));

#define N_NODES 50000
#define DIM 128

// ---------------------------------------------------------------------------
// Scatter-aggregate: agg[tgt[e]] += x[src[e]]  (sum aggregation over edges)
// One thread per (edge, 4-float chunk): 32 threads cover one 128-float row.
// L2-resident (x and agg both << 192MB L2), limited by f32 atomic rate.
// ---------------------------------------------------------------------------
__global__ __launch_bounds__(256) void gnn_scatter_add(
    const float* __restrict__ x,
    const long long* __restrict__ src,
    const long long* __restrict__ tgt,
    float* __restrict__ agg, int E)
{
    long long gid = (long long)blockIdx.x * blockDim.x + threadIdx.x;
    if (gid >= (long long)E * 32) return;
    int e     = (int)(gid >> 5);
    int chunk = (int)(gid & 31);

    int s = (int)src[e];
    int t = (int)tgt[e];

    float4 v = ((const float4*)(x + (long long)s * DIM))[chunk];
    float* out = agg + (long long)t * DIM + chunk * 4;
    unsafeAtomicAdd(out + 0, v.x);
    unsafeAtomicAdd(out + 1, v.y);
    unsafeAtomicAdd(out + 2, v.z);
    unsafeAtomicAdd(out + 3, v.w);
}

// ---------------------------------------------------------------------------
// Fused GraphConv GEMM: out = relu(agg @ Wrel^T + brel + x @ Wroot^T)
// Block = 256 threads = 8 waves; each wave owns a 16x128 output tile.
// Uses V_WMMA_F32_16X16X4_F32 (fp32 in / fp32 acc) -- matches fp32 reference.
//
// Fragment layouts (ISA 7.12.2, wave32):
//   A (16x4):  lane holds M=lane%16; reg0/reg1 = K = 2*(lane>>4) + {0,1}
//   B (4x16):  lane holds N=lane%16; reg0/reg1 = K = 2*(lane>>4) + {0,1}
//              B[k][n] = W[n][k]  -> contiguous float2 from row-major W
//   C/D (16x16): reg g holds row  g + 8*(lane>>4), col lane%16
// ---------------------------------------------------------------------------
__global__ __launch_bounds__(256) void gnn_graphconv_wmma(
    const float* __restrict__ agg,
    const float* __restrict__ xin,
    const float* __restrict__ Wrel,
    const float* __restrict__ brel,
    const float* __restrict__ Wroot,
    float* __restrict__ out)
{
    const int wave  = threadIdx.x >> 5;
    const int lane  = threadIdx.x & 31;
    const int row0  = blockIdx.x * 128 + wave * 16;
    if (row0 >= N_NODES) return;      // wave-uniform: EXEC stays all-1s

    const int n_lo  = lane & 15;
    const int khalf = (lane >> 4) * 2;     // 0 or 2
    const int m     = row0 + n_lo;         // A-fragment row for this lane

    // Accumulators initialized with the bias (bias depends only on column n).
    v8f acc[8];
#pragma unroll
    for (int nt = 0; nt < 8; ++nt) {
        float bv = brel[nt * 16 + n_lo];
        acc[nt] = (v8f){bv, bv, bv, bv, bv, bv, bv, bv};
    }

    const float* arow = agg + (long long)m * DIM;
    const float* xrow = xin + (long long)m * DIM;

    for (int kb = 0; kb < DIM; kb += 4) {
        const int k0 = kb + khalf;
        v2f aR = *(const v2f*)(arow + k0);   // agg fragment (rel path)
        v2f aX = *(const v2f*)(xrow + k0);   // x   fragment (root path)
#pragma unroll
        for (int nt = 0; nt < 8; ++nt) {
            const int n = nt * 16 + n_lo;
            v2f bR = *(const v2f*)(Wrel  + (long long)n * DIM + k0);
            v2f bX = *(const v2f*)(Wroot + (long long)n * DIM + k0);
            acc[nt] = __builtin_amdgcn_wmma_f32_16x16x4_f32(
                false, aR, false, bR, (short)0, acc[nt], false, false);
            acc[nt] = __builtin_amdgcn_wmma_f32_16x16x4_f32(
                false, aX, false, bX, (short)0, acc[nt], false, false);
        }
    }

    // ReLU + store. Reg g -> row (row0 + g + 8*(lane>>4)), col nt*16 + n_lo.
    const int mbase = row0 + ((lane >> 4) * 8);
#pragma unroll
    for (int nt = 0; nt < 8; ++nt) {
#pragma unroll
        for (int g = 0; g < 8; ++g) {
            float v = acc[nt][g];
            v = v > 0.0f ? v : 0.0f;
            out[(long long)(mbase + g) * DIM + nt * 16 + n_lo] = v;
        }
    }
}

extern "C" void kernel_launch(void* const* d_in, const int* in_sizes, int n_in,
                              void* d_out, int out_size, void* d_ws, size_t ws_size,
                              hipStream_t stream)
{
    const float*     x       = (const float*)d_in[0];
    const long long* eidx    = (const long long*)d_in[1];
    const float*     W1_rel  = (const float*)d_in[2];
    const float*     b1_rel  = (const float*)d_in[3];
    const float*     W1_root = (const float*)d_in[4];
    const float*     W2_rel  = (const float*)d_in[5];
    const float*     b2_rel  = (const float*)d_in[6];
    const float*     W2_root = (const float*)d_in[7];
    float*           out     = (float*)d_out;

    const int E = in_sizes[1] / 2;
    const long long* src = eidx;       // edge_index[0]
    const long long* tgt = eidx + E;   // edge_index[1]

    const size_t fbytes = (size_t)N_NODES * DIM * sizeof(float);
    float* agg = (float*)d_ws;
    float* h   = (float*)((char*)d_ws + fbytes);

    const long long swork   = (long long)E * 32;
    const int sblocks = (int)((swork + 255) / 256);
    const int gblocks = (N_NODES + 127) / 128;

    // ---- layer 1 ----
    hipMemsetAsync(agg, 0, fbytes, stream);
    gnn_scatter_add<<<sblocks, 256, 0, stream>>>(x, src, tgt, agg, E);
    gnn_graphconv_wmma<<<gblocks, 256, 0, stream>>>(agg, x, W1_rel, b1_rel, W1_root, h);

    // ---- layer 2 ----
    hipMemsetAsync(agg, 0, fbytes, stream);
    gnn_scatter_add<<<sblocks, 256, 0, stream>>>(h, src, tgt, agg, E);
    gnn_graphconv_wmma<<<gblocks, 256, 0, stream>>>(agg, h, W2_rel, b2_rel, W2_root, out);
}